// FrankensteinModel_53824530153643
// MI455X (gfx1250) — compile-verified
//
#include <hip/hip_runtime.h>
#include <math.h>

// ---- model constants (from reference) ----
#define B_   32
#define L_   100
#define C_   142
#define DM_  512
#define DI_  1024
#define N_   16
#define K_   4
#define DTR_ 32
#define NL_  6
#define OD_  128

typedef float v2f __attribute__((ext_vector_type(2)));
typedef float v8f __attribute__((ext_vector_type(8)));

// ---------------------------------------------------------------------------
// Input projection: h[b,l,:] = x[b,l,:] @ inp_w^T + inp_b + pos[l,:]
// K=142 not a multiple of 4 -> plain VALU dot (465 MFLOP total, negligible).
// ---------------------------------------------------------------------------
__global__ void input_proj_kernel(const float* __restrict__ x,
                                  const float* __restrict__ w,
                                  const float* __restrict__ bias,
                                  const float* __restrict__ pos,
                                  float* __restrict__ h)
{
    int idx = blockIdx.x * blockDim.x + threadIdx.x;       // (b*L + l)*DM + d
    if (idx >= B_ * L_ * DM_) return;
    int d  = idx % DM_;
    int bl = idx / DM_;
    int l  = bl % L_;
    const float* xr = x + (size_t)bl * C_;
    const float* wr = w + (size_t)d  * C_;
    float s = bias[d];
    #pragma unroll 2
    for (int c = 0; c < C_; ++c) s += xr[c] * wr[c];
    h[idx] = s + pos[(size_t)l * DM_ + d];
}

// ---------------------------------------------------------------------------
// Row LayerNorm: rows x D, with configurable row stride (for last-token LN).
// One 256-thread block per row (8 wave32's).
// ---------------------------------------------------------------------------
__global__ void layernorm_kernel(const float* __restrict__ x,
                                 const float* __restrict__ g,
                                 const float* __restrict__ b,
                                 float* __restrict__ y,
                                 int D, int rowStride)
{
    int row = blockIdx.x;
    const float* xr = x + (size_t)row * rowStride;
    float*       yr = y + (size_t)row * D;
    int tid = threadIdx.x;

    float s1 = 0.f, s2 = 0.f;
    for (int i = tid; i < D; i += blockDim.x) {
        float v = xr[i];
        s1 += v; s2 += v * v;
    }
    for (int off = 16; off >= 1; off >>= 1) {
        s1 += __shfl_xor(s1, off, 32);
        s2 += __shfl_xor(s2, off, 32);
    }
    __shared__ float ws1[8], ws2[8], mv[2];
    int w = tid >> 5;
    if ((tid & 31) == 0) { ws1[w] = s1; ws2[w] = s2; }
    __syncthreads();
    if (tid == 0) {
        float a = 0.f, c = 0.f;
        for (int j = 0; j < 8; ++j) { a += ws1[j]; c += ws2[j]; }
        float m = a / (float)D;
        mv[0] = m;
        mv[1] = rsqrtf(c / (float)D - m * m + 1e-5f);
    }
    __syncthreads();
    float m = mv[0], r = mv[1];
    for (int i = tid; i < D; i += blockDim.x)
        yr[i] = (xr[i] - m) * r * g[i] + b[i];
}

// ---------------------------------------------------------------------------
// FP32 WMMA GEMM:  C[M x N] = act( A[M x lda] (first K cols) @ W[N x K]^T + bias )
// optionally accumulated into existing C (residual add).
//
// Each wave computes a 32x32 macro-tile as a 2x2 grid of 16x16 WMMA tiles:
// per K=4 chunk, 2 A-fragment + 2 B-fragment loads feed 4
// V_WMMA_F32_16X16X4_F32 ops (1.0 WMMA/load vs 0.5 for 1 tile/wave).
// Working set is L2-resident on MI455X (192MB L2), so direct global loads.
// Requires M % 32 == 0 && N % 32 == 0 (true for every GEMM in this model).
// ---------------------------------------------------------------------------
__global__ void wmma_gemm_f32_kernel(const float* __restrict__ A,
                                     const float* __restrict__ W,
                                     const float* __restrict__ bias,
                                     float* __restrict__ C,
                                     int M, int N, int Kdim, int lda,
                                     int accumulate, int act /*0=none,1=gelu*/)
{
    int wave = (blockIdx.x * blockDim.x + threadIdx.x) >> 5;
    int lane = threadIdx.x & 31;
    int tilesN = N >> 5;                       // 32-wide macro tiles
    int tilesM = M >> 5;
    if (wave >= tilesM * tilesN) return;       // wave-uniform: EXEC stays all-1s
    int tm = wave / tilesN;
    int tn = wave % tilesN;
    int half = lane >> 4;                      // lanes 16-31 hold K+2..K+3
    int l16  = lane & 15;

    // A fragments: rows tm*32 + {0,16} + l16 ; B fragments: cols tn*32 + {0,16} + l16
    const float* Arow0 = A + (size_t)(tm * 32 + l16)      * lda;
    const float* Arow1 = A + (size_t)(tm * 32 + 16 + l16) * lda;
    const float* Wrow0 = W + (size_t)(tn * 32 + l16)      * Kdim;  // B[k][n]=W[n][k]
    const float* Wrow1 = W + (size_t)(tn * 32 + 16 + l16) * Kdim;

    v8f acc00 = {}, acc01 = {}, acc10 = {}, acc11 = {};
    for (int k0 = 0; k0 < Kdim; k0 += 4) {
        int kb = k0 + 2 * half;
        v2f a0, a1, b0, b1;
        a0.x = Arow0[kb + 0]; a0.y = Arow0[kb + 1];
        a1.x = Arow1[kb + 0]; a1.y = Arow1[kb + 1];
        b0.x = Wrow0[kb + 0]; b0.y = Wrow0[kb + 1];
        b1.x = Wrow1[kb + 0]; b1.y = Wrow1[kb + 1];
        acc00 = __builtin_amdgcn_wmma_f32_16x16x4_f32(false, a0, false, b0,
                                                      (short)0, acc00, false, false);
        acc01 = __builtin_amdgcn_wmma_f32_16x16x4_f32(false, a0, false, b1,
                                                      (short)0, acc01, false, false);
        acc10 = __builtin_amdgcn_wmma_f32_16x16x4_f32(false, a1, false, b0,
                                                      (short)0, acc10, false, false);
        acc11 = __builtin_amdgcn_wmma_f32_16x16x4_f32(false, a1, false, b1,
                                                      (short)0, acc11, false, false);
    }

    int n0 = tn * 32 + l16;
    int n1 = n0 + 16;
    float bv0 = bias ? bias[n0] : 0.0f;
    float bv1 = bias ? bias[n1] : 0.0f;
    #pragma unroll
    for (int r = 0; r < 8; ++r) {
        int m0 = tm * 32 + r + 8 * half;       // C layout: M = r + 8*(lane>=16)
        int m1 = m0 + 16;
        float v00 = acc00[r] + bv0, v01 = acc01[r] + bv1;
        float v10 = acc10[r] + bv0, v11 = acc11[r] + bv1;
        if (act == 1) {
            v00 = 0.5f * v00 * (1.0f + erff(v00 * 0.70710678118654752f));
            v01 = 0.5f * v01 * (1.0f + erff(v01 * 0.70710678118654752f));
            v10 = 0.5f * v10 * (1.0f + erff(v10 * 0.70710678118654752f));
            v11 = 0.5f * v11 * (1.0f + erff(v11 * 0.70710678118654752f));
        }
        size_t i00 = (size_t)m0 * N + n0, i01 = (size_t)m0 * N + n1;
        size_t i10 = (size_t)m1 * N + n0, i11 = (size_t)m1 * N + n1;
        if (accumulate) {
            C[i00] += v00; C[i01] += v01; C[i10] += v10; C[i11] += v11;
        } else {
            C[i00] = v00; C[i01] = v01; C[i10] = v10; C[i11] = v11;
        }
    }
}

// ---------------------------------------------------------------------------
// Causal depthwise conv (K=4) + bias + SiLU.  xp = xz[..., :DI]
// ---------------------------------------------------------------------------
__global__ void conv_silu_kernel(const float* __restrict__ xz,
                                 const float* __restrict__ cw,
                                 const float* __restrict__ cb,
                                 float* __restrict__ xc)
{
    int idx = blockIdx.x * blockDim.x + threadIdx.x;   // (b*L + l)*DI + d
    if (idx >= B_ * L_ * DI_) return;
    int d  = idx % DI_;
    int bl = idx / DI_;
    int l  = bl % L_;
    float s = cb[d];
    #pragma unroll
    for (int k = 0; k < K_; ++k) {
        int li = l - (K_ - 1) + k;
        if (li >= 0)
            s += xz[(size_t)(bl - l + li) * (2 * DI_) + d] * cw[d * K_ + k];
    }
    xc[idx] = s / (1.0f + __expf(-s));
}

// ---------------------------------------------------------------------------
// Selective scan (sequential over L) + skip + SiLU(z) gate.
// One thread per (b, d); 16 states in registers; B/C rows are wave-uniform
// loads (hardware broadcast). dtraw already contains dt_b bias from the GEMM.
// ---------------------------------------------------------------------------
__global__ void scan_kernel(const float* __restrict__ dtraw,
                            const float* __restrict__ xc,
                            const float* __restrict__ xz,   // z at [..., DI:]
                            const float* __restrict__ xd,   // Bm at 32..47, Cm at 48..63
                            const float* __restrict__ A_log,
                            const float* __restrict__ Dp,
                            float* __restrict__ yg)
{
    int idx = blockIdx.x * blockDim.x + threadIdx.x;   // b*DI + d
    if (idx >= B_ * DI_) return;
    int d = idx % DI_;
    int b = idx / DI_;

    float An[N_];
    #pragma unroll
    for (int n = 0; n < N_; ++n) An[n] = -__expf(A_log[d * N_ + n]);
    float st[N_];
    #pragma unroll
    for (int n = 0; n < N_; ++n) st[n] = 0.0f;
    float Dpd = Dp[d];

    for (int l = 0; l < L_; ++l) {
        size_t bl = (size_t)b * L_ + l;
        float raw = dtraw[bl * DI_ + d];
        float dtv = (raw > 20.0f) ? raw : log1pf(__expf(raw));   // softplus
        float xcv = xc[bl * DI_ + d];
        float zv  = xz[bl * (2 * DI_) + DI_ + d];
        const float* xdr = xd + bl * (DTR_ + 2 * N_);
        float dtx = dtv * xcv;
        float y = 0.0f;
        #pragma unroll
        for (int n = 0; n < N_; ++n) {
            float a = __expf(dtv * An[n]);
            st[n] = a * st[n] + dtx * xdr[DTR_ + n];       // Bm
            y += st[n] * xdr[DTR_ + N_ + n];               // Cm
        }
        y += Dpd * xcv;
        float sz = zv / (1.0f + __expf(-zv));
        yg[bl * DI_ + d] = y * sz;
    }
}

// ---------------------------------------------------------------------------
extern "C" void kernel_launch(void* const* d_in, const int* in_sizes, int n_in,
                              void* d_out, int out_size, void* d_ws, size_t ws_size,
                              hipStream_t stream)
{
    const float* x      = (const float*)d_in[0];
    const float* inp_w  = (const float*)d_in[1];
    const float* inp_b  = (const float*)d_in[2];
    const float* pos    = (const float*)d_in[3];
    const float* norm_g = (const float*)d_in[4];
    const float* norm_b = (const float*)d_in[5];
    const float* in_w   = (const float*)d_in[6];
    const float* conv_w = (const float*)d_in[7];
    const float* conv_b = (const float*)d_in[8];
    const float* xproj_w= (const float*)d_in[9];
    const float* dt_w   = (const float*)d_in[10];
    const float* dt_b   = (const float*)d_in[11];
    const float* A_log  = (const float*)d_in[12];
    const float* Dp     = (const float*)d_in[13];
    const float* out_w  = (const float*)d_in[14];
    const float* lnf_g  = (const float*)d_in[15];
    const float* lnf_b  = (const float*)d_in[16];
    const float* op1_w  = (const float*)d_in[17];
    const float* op1_b  = (const float*)d_in[18];
    const float* op2_w  = (const float*)d_in[19];
    const float* op2_b  = (const float*)d_in[20];
    float* out = (float*)d_out;

    // workspace layout (floats)
    float* ws     = (float*)d_ws;
    const size_t R = (size_t)B_ * L_;            // 3200 rows
    float* h      = ws;                          // R*DM
    float* ln     = h      + R * DM_;            // R*DM
    float* xz     = ln     + R * DM_;            // R*2*DI
    float* xc     = xz     + R * 2 * DI_;        // R*DI
    float* xd     = xc     + R * DI_;            // R*64
    float* dtraw  = xd     + R * (DTR_ + 2*N_);  // R*DI
    float* yg     = dtraw  + R * DI_;            // R*DI
    float* lastln = yg     + R * DI_;            // B*DM
    float* hid    = lastln + (size_t)B_ * DM_;   // B*DM

    auto gemm = [&](const float* A, const float* W, const float* bias, float* Cm,
                    int M, int N, int Kd, int lda, int accum, int act) {
        int tiles  = (M >> 5) * (N >> 5);        // 32x32 macro tiles, 1 per wave
        int blocks = (tiles + 7) / 8;            // 8 waves / 256-thread block
        wmma_gemm_f32_kernel<<<blocks, 256, 0, stream>>>(A, W, bias, Cm,
                                                         M, N, Kd, lda, accum, act);
    };

    // 1. input projection + positional embedding
    {
        int n = B_ * L_ * DM_;
        input_proj_kernel<<<(n + 255) / 256, 256, 0, stream>>>(x, inp_w, inp_b, pos, h);
    }

    // 2. six S6 layers
    for (int i = 0; i < NL_; ++i) {
        layernorm_kernel<<<(int)R, 256, 0, stream>>>(h, norm_g + i * DM_,
                                                     norm_b + i * DM_, ln, DM_, DM_);
        // xz = ln @ in_w^T : 3200x512 @ (2048x512)^T
        gemm(ln, in_w + (size_t)i * 2 * DI_ * DM_, nullptr, xz,
             (int)R, 2 * DI_, DM_, DM_, 0, 0);
        // causal depthwise conv + SiLU
        {
            int n = B_ * L_ * DI_;
            conv_silu_kernel<<<(n + 255) / 256, 256, 0, stream>>>(
                xz, conv_w + (size_t)i * DI_ * K_, conv_b + (size_t)i * DI_, xc);
        }
        // xd = xc @ xproj_w^T : 3200x1024 @ (64x1024)^T
        gemm(xc, xproj_w + (size_t)i * (DTR_ + 2*N_) * DI_, nullptr, xd,
             (int)R, DTR_ + 2 * N_, DI_, DI_, 0, 0);
        // dtraw = xd[:, :32] @ dt_w^T + dt_b : 3200x32 @ (1024x32)^T
        gemm(xd, dt_w + (size_t)i * DI_ * DTR_, dt_b + (size_t)i * DI_, dtraw,
             (int)R, DI_, DTR_, DTR_ + 2 * N_, 0, 0);
        // sequential selective scan + gating
        {
            int n = B_ * DI_;
            scan_kernel<<<(n + 255) / 256, 256, 0, stream>>>(
                dtraw, xc, xz, xd, A_log + (size_t)i * DI_ * N_,
                Dp + (size_t)i * DI_, yg);
        }
        // h += yg @ out_w^T : 3200x1024 @ (512x1024)^T, residual accumulate
        gemm(yg, out_w + (size_t)i * DM_ * DI_, nullptr, h,
             (int)R, DM_, DI_, DI_, 1, 0);
    }

    // 3. final LN on last token only (row b*L + L-1), then MLP head
    layernorm_kernel<<<B_, 256, 0, stream>>>(h + (size_t)(L_ - 1) * DM_,
                                             lnf_g, lnf_b, lastln, DM_, L_ * DM_);
    gemm(lastln, op1_w, op1_b, hid, B_, DM_, DM_, DM_, 0, /*gelu*/1);
    gemm(hid, op2_w, op2_b, out, B_, OD_, DM_, DM_, 0, 0);
}